// LinearStringEncoder_91199335563328
// MI455X (gfx1250) — compile-verified
//
#include <hip/hip_runtime.h>

// Problem constants from the reference: B=4096, L=200, V=50000, H=128
#define V_SZ 50000
#define H_SZ 128
#define L_SZ 200
#define LPAD 204     // per-scene LDS token stride (4 zero pads, keeps int2 aligned)
#define SCN  16      // scenes per block (16 = WMMA N dimension)

typedef float v2f __attribute__((ext_vector_type(2)));
typedef float v8f __attribute__((ext_vector_type(8)));

// ---------------------------------------------------------------------------
// Pass 1: W[H,V] -> Wt[V,H] so a token's 128-float feature row is contiguous.
// ---------------------------------------------------------------------------
__global__ __launch_bounds__(256) void wt_transpose(const float* __restrict__ W,
                                                    float* __restrict__ Wt) {
  __shared__ float tile[32][33];
  const int tx = threadIdx.x;            // 0..31
  const int ty = threadIdx.y;            // 0..7
  const int v0 = blockIdx.x * 32;
  const int h0 = blockIdx.y * 32;        // H=128 -> gridDim.y = 4, always in range
#pragma unroll
  for (int i = 0; i < 4; ++i) {
    const int h = h0 + ty + i * 8;
    const int v = v0 + tx;
    if (v < V_SZ) tile[ty + i * 8][tx] = W[(size_t)h * V_SZ + v];
  }
  __syncthreads();
#pragma unroll
  for (int i = 0; i < 4; ++i) {
    const int v = v0 + ty + i * 8;
    const int h = h0 + tx;
    if (v < V_SZ) Wt[(size_t)v * H_SZ + h] = tile[tx][ty + i * 8];
  }
}

// ---------------------------------------------------------------------------
// Pass 2: out[i,h] = b[h] + sum_{j<len_i} Wt[tok_{i,j}, h]
// Block = 16 scenes x 256 threads (8 waves). Wave w owns h in [16w, 16w+16).
// Per 4-token group: A = gathered W rows (16x4), B = scene-selection mask
// (4x16), accumulated with v_wmma_f32_16x16x4_f32 into a 16x16 f32 tile.
// TR selects contiguous gathers from Wt (fast) vs strided gathers from W.
// ---------------------------------------------------------------------------
template <bool TR>
__global__ __launch_bounds__(256) void encoder(const int* __restrict__ tokens,
                                               const int* __restrict__ lengths,
                                               const float* __restrict__ W,
                                               const float* __restrict__ bias,
                                               const float* __restrict__ Wt,
                                               float* __restrict__ out) {
  __shared__ int toks[SCN * LPAD];       // ~13 KB
  __shared__ int lens[SCN];

  const int tid        = threadIdx.x;
  const int scene_base = blockIdx.x * SCN;

  // Stage this block's tokens + lengths into LDS (zero the 4 pad slots).
  for (int idx = tid; idx < SCN * L_SZ; idx += 256) {
    const int s = idx / L_SZ;
    const int j = idx - s * L_SZ;
    toks[s * LPAD + j] = tokens[(size_t)(scene_base + s) * L_SZ + j];
  }
  if (tid < SCN * 4) toks[(tid >> 2) * LPAD + L_SZ + (tid & 3)] = 0;
  if (tid < SCN) {
    int l = lengths[scene_base + tid];
    lens[tid] = l < 0 ? 0 : (l > L_SZ ? L_SZ : l);
  }
  __syncthreads();

  const int wave   = tid >> 5;           // 0..7
  const int lane   = tid & 31;
  const int half   = lane >> 4;          // 0 or 1
  const int n      = lane & 15;          // A: row m; B/D: column n (scene)
  const int h_base = wave * 16;

  const float* __restrict__ src = TR ? Wt : W;
  // Per-lane constant part of the gather index.
  const size_t off = TR ? (size_t)(h_base + n)
                        : (size_t)(h_base + n) * V_SZ;
  auto gidx = [&](int t) -> size_t {
    return TR ? (((size_t)(unsigned)t << 7) + off)   // t*H + h
              : ((size_t)(unsigned)t + off);         // h*V + t
  };

  v8f acc = {};                          // 16x16 f32 accumulator tile

  for (int s = 0; s < SCN; ++s) {
    const int len = lens[s];             // block-uniform -> EXEC stays all-ones
    if (len <= 0) continue;
    const float sel  = (n == s) ? 1.0f : 0.0f;   // selection column, loop-invariant
    const int   base = s * LPAD + 2 * half;      // lanes 0-15: K=0,1; 16-31: K=2,3
    const int   len4 = len & ~3;

    // Software pipeline: tt always holds the token pair for group j.
    int2 tt = *(const int2*)&toks[base];
    int  j  = 0;
    for (; j < len4; j += 4) {
      const int2 cur = tt;
      tt = *(const int2*)&toks[base + j + 4];    // prefetch next (pads make it safe)
      const float a0 = src[gidx(cur.x)];
      const float a1 = src[gidx(cur.y)];
      v2f A  = {a0, a1};
      v2f Bm = {sel, sel};
      acc = __builtin_amdgcn_wmma_f32_16x16x4_f32(
          false, A, false, Bm, (short)0, acc, false, false);
    }
    if (j < len) {                       // ragged tail group (uniform branch)
      const int2 cur = tt;
      const int  j0  = j + 2 * half;
      const float a0 = src[gidx(cur.x)];
      const float a1 = src[gidx(cur.y)];
      v2f A  = {a0, a1};
      v2f Bm = {(j0 < len) ? sel : 0.0f, (j0 + 1 < len) ? sel : 0.0f};
      acc = __builtin_amdgcn_wmma_f32_16x16x4_f32(
          false, A, false, Bm, (short)0, acc, false, false);
    }
  }

  // D layout: VGPR r -> (m = r + 8*half, n = lane&15)
  const int scene = scene_base + n;
#pragma unroll
  for (int r = 0; r < 8; ++r) {
    const int h = h_base + r + 8 * half;
    out[(size_t)scene * H_SZ + h] = acc[r] + bias[h];
  }
}

// ---------------------------------------------------------------------------
extern "C" void kernel_launch(void* const* d_in, const int* in_sizes, int n_in,
                              void* d_out, int out_size, void* d_ws, size_t ws_size,
                              hipStream_t stream) {
  const int*   tokens  = (const int*)d_in[0];
  const int*   lengths = (const int*)d_in[1];
  const float* W       = (const float*)d_in[2];
  const float* bias    = (const float*)d_in[3];
  float*       out     = (float*)d_out;
  float*       Wt      = (float*)d_ws;

  const int B = in_sizes[1];             // 4096 scenes
  const size_t wt_bytes = (size_t)V_SZ * H_SZ * sizeof(float);   // 25.6 MB
  const bool use_t = ws_size >= wt_bytes;

  if (use_t) {
    dim3 tgrid((V_SZ + 31) / 32, H_SZ / 32);
    dim3 tblk(32, 8);
    wt_transpose<<<tgrid, tblk, 0, stream>>>(W, Wt);
    encoder<true><<<B / SCN, 256, 0, stream>>>(tokens, lengths, W, bias, Wt, out);
  } else {
    encoder<false><<<B / SCN, 256, 0, stream>>>(tokens, lengths, W, bias, Wt, out);
  }
}